// FFT1D_18382460027426
// MI455X (gfx1250) — compile-verified
//
#include <hip/hip_runtime.h>

// "FFT with per-stage magnitude" — real-valued butterflies, two fused LDS passes.
// N = 2^22, 22 stages.
//  Pass 1: stages 1..12. 4096-elem chunks; bit-reversed per-lane async gather
//          into LDS (ASYNCcnt), butterflies in LDS, TDM store of the chunk.
//  Pass 2: stages 13..22. 1024 elems at stride 4096 (fixed low-12 bits),
//          TDM 2D strided load/store (TENSORcnt), in-place on d_out.

#define NTOT        (1u << 22)
#define TOTAL_BITS  22u
#define S1_LOG      12
#define CH1         (1u << S1_LOG)   // 4096 elements / block, pass 1
#define S2_LOG      10
#define CH2         (1u << S2_LOG)   // 1024 elements / block, pass 2
#define TPB         256              // 8 wave32 waves
#define PI_F        3.14159265358979323846f

typedef unsigned int u32;
typedef __attribute__((ext_vector_type(4))) u32 u32x4;
typedef __attribute__((ext_vector_type(8))) u32 u32x8;

// Low 32 bits of a flat shared pointer == LDS byte offset (aperture base has
// zero low bits; the hardware uses only the low 32 address bits).
__device__ __forceinline__ unsigned lds_off(const void* p) {
  return (unsigned)(unsigned long long)p;
}

// ---- per-lane async global->LDS (ASYNCcnt) -------------------------------
__device__ __forceinline__ void async_g2l_b32(const float* g, const float* l) {
  asm volatile("global_load_async_to_lds_b32 %0, %1, off"
               :: "v"(lds_off(l)), "v"(g)
               : "memory");
}
__device__ __forceinline__ void wait_async0() {
  asm volatile("s_wait_asynccnt 0x0" ::: "memory");
}

// ---- Tensor Data Mover (TENSORcnt) ---------------------------------------
// D# group 0: [1:0]=count=1, [63:32]=lds_addr, [120:64]=global_addr, [127:126]=type=2
__device__ __forceinline__ u32x4 tdm_g0(const void* gptr, unsigned lds_byte) {
  const unsigned long long ga = (unsigned long long)gptr;
  u32x4 g;
  g[0] = 1u;
  g[1] = lds_byte;
  g[2] = (u32)ga;
  g[3] = ((u32)(ga >> 32) & 0x01FFFFFFu) | (2u << 30);
  return g;
}
// D# group 1: data_size=4B; 2D tensor/tile; dim0 stride in elements.
__device__ __forceinline__ u32x8 tdm_g1(u32 dim0, u32 dim1, u32 tile0, u32 tile1,
                                        unsigned long long stride0) {
  u32x8 g;
  g[0] = 2u << 16;                                   // data_size=2 (4 bytes)
  g[1] = (dim0 & 0xFFFFu) << 16;                     // tensor_dim0[15:0] @63:48
  g[2] = (dim0 >> 16) | ((dim1 & 0xFFFFu) << 16);    // dim0 hi, dim1 lo
  g[3] = (dim1 >> 16) | (tile0 << 16);               // dim1 hi, tile_dim0 @127:112
  g[4] = tile1;                                      // tile_dim1; tile_dim2=0
  g[5] = (u32)stride0;                               // tensor_dim0_stride lo32
  g[6] = (u32)(stride0 >> 32) & 0xFFFFu;             // stride hi16; dim1_stride=0
  g[7] = 0u;
  return g;
}
__device__ __forceinline__ void tdm_load(u32x4 g0, u32x8 g1) {
  asm volatile("tensor_load_to_lds %0, %1" :: "s"(g0), "s"(g1) : "memory");
}
__device__ __forceinline__ void tdm_store(u32x4 g0, u32x8 g1) {
  asm volatile("tensor_store_from_lds %0, %1" :: "s"(g0), "s"(g1) : "memory");
}
__device__ __forceinline__ void wait_tensor0() {
  __builtin_amdgcn_s_wait_tensorcnt(0);
}

// ---- real->magnitude butterfly ------------------------------------------
__device__ __forceinline__ void mag_butterfly(float* __restrict__ sh,
                                              int ie, int io, float ang) {
  const float e = sh[ie];
  const float o = sh[io];
  float sn, c;
  __sincosf(ang, &sn, &c);
  const float tr = c * o;
  const float ti = sn * o;
  const float ap = e + tr;
  const float am = e - tr;
  sh[ie] = sqrtf(ap * ap + ti * ti);
  sh[io] = sqrtf(am * am + ti * ti);
}

// ---- Pass 1: bit-reversed gather + stages 1..12 in LDS -------------------
__global__ __launch_bounds__(TPB) void fft_mag_local(const float* __restrict__ x,
                                                     float* __restrict__ y) {
  __shared__ float sh[CH1];
  const unsigned tid  = threadIdx.x;
  const unsigned base = (unsigned)blockIdx.x << S1_LOG;

  // sh[j] = x[bitrev22(base + j)] — per-lane async DMA straight into LDS.
  #pragma unroll
  for (int it = 0; it < (int)(CH1 / TPB); ++it) {
    const unsigned j = tid + (unsigned)it * TPB;
    const unsigned src = __brev(base + j) >> (32u - TOTAL_BITS);
    async_g2l_b32(x + src, &sh[j]);
  }
  wait_async0();
  __syncthreads();

  for (int s = 1; s <= S1_LOG; ++s) {
    const int half = 1 << (s - 1);
    #pragma unroll
    for (int it = 0; it < (int)(CH1 / 2 / TPB); ++it) {
      const int b  = (int)tid + it * TPB;
      const int j  = b & (half - 1);
      const int ie = ((b >> (s - 1)) << s) + j;
      mag_butterfly(sh, ie, ie + half, -PI_F * (float)j / (float)half);
    }
    __syncthreads();
  }

  // Contiguous 16KB chunk out: one TDM store issued by wave 0.
  if (tid < 32) {
    tdm_store(tdm_g0(y + base, lds_off(sh)),
              tdm_g1(/*dim0*/CH1, /*dim1*/1, /*tile0*/CH1, /*tile1*/1, /*stride0*/CH1));
    wait_tensor0();   // hold LDS alive until the DMA drains
  }
}

// ---- Pass 2: stages 13..22, elements at stride 4096, fixed low-12 bits ---
__global__ __launch_bounds__(TPB) void fft_mag_global(float* __restrict__ y) {
  __shared__ float sh[CH2];
  const unsigned tid = threadIdx.x;
  const unsigned low = blockIdx.x;                    // 0..4095
  const float lowfrac = (float)low * (1.0f / (float)CH1);

  // sh[l] = y[low + l*4096] — one 2D strided TDM load (tile 1 x 1024).
  if (tid < 32) {
    tdm_load(tdm_g0(y + low, lds_off(sh)),
             tdm_g1(/*dim0*/CH1, /*dim1*/CH2, /*tile0*/1, /*tile1*/CH2, /*stride0*/CH1));
    wait_tensor0();
  }
  __syncthreads();

  for (int s = 1; s <= S2_LOG; ++s) {                 // global stage = s + 12
    const int half = 1 << (s - 1);
    #pragma unroll
    for (int it = 0; it < (int)(CH2 / 2 / TPB); ++it) {
      const int b  = (int)tid + it * TPB;
      const int j  = b & (half - 1);
      const int ie = ((b >> (s - 1)) << s) + j;
      // global twiddle: j_g = j*4096 + low, m_g = 2*half*4096
      mag_butterfly(sh, ie, ie + half,
                    -PI_F * ((float)j + lowfrac) / (float)half);
    }
    __syncthreads();
  }

  // Strided writeback: one 2D TDM store.
  if (tid < 32) {
    tdm_store(tdm_g0(y + low, lds_off(sh)),
              tdm_g1(CH1, CH2, 1, CH2, CH1));
    wait_tensor0();
  }
}

extern "C" void kernel_launch(void* const* d_in, const int* in_sizes, int n_in,
                              void* d_out, int out_size, void* d_ws, size_t ws_size,
                              hipStream_t stream) {
  (void)in_sizes; (void)n_in; (void)out_size; (void)d_ws; (void)ws_size;
  const float* x = (const float*)d_in[0];
  float*       y = (float*)d_out;

  fft_mag_local <<<NTOT / CH1, TPB, 0, stream>>>(x, y);   // 1024 blocks
  fft_mag_global<<<CH1,        TPB, 0, stream>>>(y);      // 4096 blocks
}